// AoERouter_11184094839570
// MI455X (gfx1250) — compile-verified
//
#include <hip/hip_runtime.h>
#include <hip/hip_bf16.h>
#include <math.h>

typedef float v2f __attribute__((ext_vector_type(2)));
typedef float v8f __attribute__((ext_vector_type(8)));

#define D_MODEL 1024
#define N_EXP   32
#define D_LOW   64
#define NFEAT   (N_EXP * D_LOW)   // 2048
#define NTOK    1024
#define TOPK    2

// =====================================================================
// Kernel 1: feats[N, E*dl] = x[N, D] @ w_down[E*dl, D]^T   (fp32 WMMA)
// 4 waves/block (2x2), each wave owns a 64x64 tile = 4x4 grid of
// v_wmma_f32_16x16x4_f32 accumulators -> 128 B of fresh A/B fragment
// per WMMA (4x reuse of each fragment), half the operand traffic of a
// 2x2 tile. A 16x4 f32 fragment (ISA 7.12.2): lane%16 = M row,
// lane/16 selects K pair {0,1}/{2,3} -> one 8B load per fragment; a
// column of B is a contiguous row of w_down so B uses the same pattern.
// =====================================================================
__global__ __launch_bounds__(128)
void gemm_down_kernel(const float* __restrict__ x,
                      const float* __restrict__ w,
                      float* __restrict__ feats) {
  const int lane = threadIdx.x & 31;
  const int wave = threadIdx.x >> 5;
  const int wm = wave & 1;            // 2 waves along M
  const int wn = wave >> 1;           // 2 waves along N
  const int m_wave = blockIdx.x * 128 + wm * 64;
  const int n_wave = blockIdx.y * 128 + wn * 64;
  const int lr = lane & 15;           // row (A) / col (B) within 16
  const int kh = (lane >> 4) << 1;    // K sub-offset: 0 or 2

  const float* pa[4];
  const float* pb[4];
#pragma unroll
  for (int i = 0; i < 4; ++i) {
    pa[i] = x + (size_t)(m_wave + i * 16 + lr) * D_MODEL + kh;
    pb[i] = w + (size_t)(n_wave + i * 16 + lr) * D_MODEL + kh;
  }

  v8f acc[4][4];
#pragma unroll
  for (int i = 0; i < 4; ++i)
#pragma unroll
    for (int j = 0; j < 4; ++j) acc[i][j] = (v8f){};

#pragma unroll 4
  for (int k = 0; k < D_MODEL; k += 4) {
    v2f a[4], b[4];
#pragma unroll
    for (int i = 0; i < 4; ++i) {
      a[i] = *(const v2f*)(pa[i] + k);
      b[i] = *(const v2f*)(pb[i] + k);
    }
#pragma unroll
    for (int i = 0; i < 4; ++i)
#pragma unroll
      for (int j = 0; j < 4; ++j)
        acc[i][j] = __builtin_amdgcn_wmma_f32_16x16x4_f32(
            false, a[i], false, b[j], (short)0, acc[i][j], false, false);
  }

  // C/D 16x16 f32 layout: VGPR r -> M = r (lanes 0-15) / M = 8+r
  // (lanes 16-31), N = lane%16.
  const int cn = lane & 15;
  const int mr = (lane >> 4) << 3;
#pragma unroll
  for (int i = 0; i < 4; ++i) {
#pragma unroll
    for (int j = 0; j < 4; ++j) {
#pragma unroll
      for (int r = 0; r < 8; ++r) {
        feats[(size_t)(m_wave + i * 16 + mr + r) * NFEAT
              + (n_wave + j * 16 + cn)] = acc[i][j][r];
      }
    }
  }
}

// =====================================================================
// Kernel 2: router — one wave32 per token, lane e = expert e.
// logits -> softmax (wave shuffles) -> top-2 with jax tie-break
// (lower index wins) -> renormalized weights. probs stored for aux loss.
// =====================================================================
__global__ __launch_bounds__(32)
void router_kernel(const float* __restrict__ feats,
                   const float* __restrict__ router_w,
                   float* __restrict__ probs,
                   int* __restrict__ topk_i,
                   float* __restrict__ topk_w) {
  const int n = blockIdx.x;
  const int e = threadIdx.x;  // 0..31 == expert id
  const float* f = feats + (size_t)n * NFEAT + e * D_LOW;
  float logit = 0.f;
#pragma unroll 8
  for (int l = 0; l < D_LOW; ++l) logit += f[l] * router_w[l];

  // softmax over the 32 lanes
  float m = logit;
  for (int off = 16; off > 0; off >>= 1) m = fmaxf(m, __shfl_xor(m, off, 32));
  float ex = expf(logit - m);
  float s = ex;
  for (int off = 16; off > 0; off >>= 1) s += __shfl_xor(s, off, 32);
  const float p = ex / s;
  probs[(size_t)n * N_EXP + e] = p;

  // top-1 (ties -> lower expert index, matching jax.lax.top_k)
  float v1 = p; int i1 = e;
  for (int off = 16; off > 0; off >>= 1) {
    float ov = __shfl_xor(v1, off, 32);
    int   oi = __shfl_xor(i1, off, 32);
    if (ov > v1 || (ov == v1 && oi < i1)) { v1 = ov; i1 = oi; }
  }
  // top-2
  float v2 = (e == i1) ? -1.f : p; int i2 = e;
  for (int off = 16; off > 0; off >>= 1) {
    float ov = __shfl_xor(v2, off, 32);
    int   oi = __shfl_xor(i2, off, 32);
    if (ov > v2 || (ov == v2 && oi < i2)) { v2 = ov; i2 = oi; }
  }
  if (e == 0) {
    const float inv = 1.f / (v1 + v2);
    topk_i[n * 2 + 0] = i1;
    topk_i[n * 2 + 1] = i2;
    topk_w[n * 2 + 0] = v1 * inv;
    topk_w[n * 2 + 1] = v2 * inv;
  }
}

// =====================================================================
// Kernel 3: per-token gathered up-proj.
// out[n,:] = sum_k w_k * gelu(feats[n, e_k*64 : +64]) @ w_up[e_k]   (64x1024)
// Block = one token, 256 threads, 4 outputs each via float4 loads of w_up
// (w_up is 8 MB -> L2 resident, reads coalesced 4 KB per l-step).
// =====================================================================
__global__ __launch_bounds__(256)
void upproj_kernel(const float* __restrict__ feats,
                   const float* __restrict__ w_up,
                   const int* __restrict__ topk_i,
                   const float* __restrict__ topk_w,
                   float* __restrict__ out) {
  __shared__ float s_act[TOPK][D_LOW];
  __shared__ int   s_e[TOPK];
  const int n = blockIdx.x;
  const int t = threadIdx.x;

  if (t < TOPK) s_e[t] = topk_i[n * 2 + t];
  if (t < TOPK * D_LOW) {
    const int k = t / D_LOW, l = t % D_LOW;
    const int e = topk_i[n * 2 + k];
    const float v = feats[(size_t)n * NFEAT + e * D_LOW + l];
    const float g = 0.5f * v * (1.f + erff(v * 0.70710678118654752440f)); // exact GELU
    s_act[k][l] = g * topk_w[n * 2 + k]; // fold routing weight into activation
  }
  __syncthreads();

  const int d0 = t * 4;
  float4 acc = make_float4(0.f, 0.f, 0.f, 0.f);
#pragma unroll
  for (int k = 0; k < TOPK; ++k) {
    const float* W = w_up + (size_t)s_e[k] * D_LOW * D_MODEL + d0;
#pragma unroll 8
    for (int l = 0; l < D_LOW; ++l) {
      const float a = s_act[k][l];
      const float4 w4 = *(const float4*)(W + (size_t)l * D_MODEL);
      acc.x += a * w4.x; acc.y += a * w4.y;
      acc.z += a * w4.z; acc.w += a * w4.w;
    }
  }
  *(float4*)(out + (size_t)n * D_MODEL + d0) = acc;
}

// =====================================================================
// Kernel 4: aux load-balancing loss — deterministic fixed-order reduction
// (no float atomics). 256 threads: expert e = t/8 over 8 token-slots.
// =====================================================================
__global__ __launch_bounds__(256)
void aux_kernel(const float* __restrict__ probs,
                const int* __restrict__ topk_i,
                float* __restrict__ aux_out) {
  __shared__ float sp[256];
  __shared__ float sl[256];
  __shared__ float terms[N_EXP];
  const int t = threadIdx.x;
  const int e = t >> 3;      // 0..31
  const int slot = t & 7;    // 0..7

  float ap = 0.f, al = 0.f;
  for (int n = slot; n < NTOK; n += 8) {
    ap += probs[(size_t)n * N_EXP + e];
    al += (topk_i[n * 2] == e ? 1.f : 0.f) + (topk_i[n * 2 + 1] == e ? 1.f : 0.f);
  }
  sp[t] = ap; sl[t] = al;
  __syncthreads();
  for (int off = 4; off > 0; off >>= 1) {
    if (slot < off) { sp[t] += sp[t + off]; sl[t] += sl[t + off]; }
    __syncthreads();
  }
  if (slot == 0)
    terms[e] = (float)N_EXP * (sp[t] / (float)NTOK) * (sl[t] / (float)NTOK);
  __syncthreads();
  if (t == 0) {
    float a = 0.f;
    for (int i = 0; i < N_EXP; ++i) a += terms[i];
    *aux_out = a;
  }
}

// =====================================================================
extern "C" void kernel_launch(void* const* d_in, const int* in_sizes, int n_in,
                              void* d_out, int out_size, void* d_ws, size_t ws_size,
                              hipStream_t stream) {
  (void)in_sizes; (void)n_in; (void)out_size; (void)ws_size;

  const float* x        = (const float*)d_in[0];  // [1,1024,1024]
  const float* w_down   = (const float*)d_in[1];  // [2048,1024]
  const float* router_w = (const float*)d_in[2];  // [1,64]
  const float* w_up     = (const float*)d_in[3];  // [32,64,1024]
  // d_in[4] = topk scalar (compile-time TOPK=2)

  float* out = (float*)d_out;                      // out [1024*1024] ++ aux [1]

  char*  ws     = (char*)d_ws;
  float* feats  = (float*)ws;                                      // 8 MB
  float* probs  = (float*)(ws + (size_t)NTOK * NFEAT * 4);         // 128 KB
  int*   topk_i = (int*)  (ws + (size_t)NTOK * NFEAT * 4
                              + (size_t)NTOK * N_EXP * 4);         // 8 KB
  float* topk_w = (float*)((char*)topk_i + (size_t)NTOK * TOPK * 4);

  gemm_down_kernel<<<dim3(NTOK / 128, NFEAT / 128), 128, 0, stream>>>(x, w_down, feats);
  router_kernel  <<<NTOK, 32,  0, stream>>>(feats, router_w, probs, topk_i, topk_w);
  upproj_kernel  <<<NTOK, 256, 0, stream>>>(feats, w_up, topk_i, topk_w, out);
  aux_kernel     <<<1,    256, 0, stream>>>(probs, topk_i, out + (size_t)NTOK * D_MODEL);
}